// UpwardPropagationCell_53360673685857
// MI455X (gfx1250) — compile-verified
//
#include <hip/hip_runtime.h>
#include <cstdint>
#include <cstddef>

// ---------------------------------------------------------------------------
// UpwardPropagationCell scan: B x C independent length-L recurrences.
// One wave32 per batch row b, lane = channel c (C == 32).
// Input staged via CDNA5 async global->LDS copies (ASYNCcnt pipeline),
// output streamed with nontemporal 128-bit stores.
// ---------------------------------------------------------------------------

typedef float v4f __attribute__((ext_vector_type(4)));
typedef int   v4i __attribute__((vector_size(16)));   // spelling the builtin expects
typedef __attribute__((address_space(1))) v4i* global_v4i_ptr;
typedef __attribute__((address_space(3))) v4i* lds_v4i_ptr;

#define RING_D          6      // LDS ring depth in steps (2 async B128 per step)
#define WAVES_PER_BLOCK 8
#define STEP_FLOATS     256    // C(32) * 8 floats per (b, l)

// s_wait_asynccnt <= n : done-signals retire in issue order, so waiting to
// 2*(RING_D-1) guarantees the oldest step's two copies have landed in LDS.
#define WAIT_ASYNCCNT(n) asm volatile("s_wait_asynccnt %0" :: "n"(n) : "memory")
// Ensure our ds_loads have drained into VGPRs before the async engine is
// allowed to overwrite the slot (WAR hazard: DS and ASYNC are unordered).
#define WAIT_DSCNT0()    asm volatile("s_wait_dscnt 0" ::: "memory")

__device__ __forceinline__ void async_copy_b128(const float* g, float* l) {
#if __has_builtin(__builtin_amdgcn_global_load_async_to_lds_b128)
  __builtin_amdgcn_global_load_async_to_lds_b128(
      (global_v4i_ptr)g,
      (lds_v4i_ptr)l,
      0, 0);
#else
  unsigned lofs = (unsigned)(uintptr_t)l;  // low 32 bits of generic LDS ptr = LDS offset
  asm volatile("global_load_async_to_lds_b128 %0, %1, off"
               :: "v"(lofs), "v"(g) : "memory");
#endif
}

__device__ __forceinline__ void cell_step(const v4f i0, const v4f i1,
                                          float& r_bot_dir, float& r_bot_dif,
                                          float& a_bot_dir, float& a_bot_dif,
                                          v4f& o0, v4f& o1) {
  const float t_dir   = i0.x, t_dif   = i0.y;
  const float e_t_dir = i0.z, e_r_dir = i0.w, e_a_dir = i1.x;
  const float e_t_dif = i1.y, e_r_dif = i1.z, e_a_dif = i1.w;

  const float e_dir = 1.0f - t_dir;
  const float e_dif = 1.0f - t_dif;
  const float d = 1.0f / (1.0f - e_dif * e_r_dif * r_bot_dif + 1e-6f);

  const float t_multi_dir     = t_dir * r_bot_dir * e_dif * e_r_dif * d + e_dir * e_t_dir * d;
  const float a_bot_multi_dir = t_dir * a_bot_dir + t_multi_dir * a_bot_dif;
  const float r_bot_multi_dir = t_dir * r_bot_dir * d + e_dir * e_t_dir * r_bot_dif * d;
  const float a_top_multi_dir = e_dir * e_a_dir + r_bot_multi_dir * e_dif * e_a_dif;
  const float r_multi_dir     = e_dir * e_r_dir + r_bot_multi_dir * (t_dif + e_dif * e_t_dif);

  const float t_multi_dif     = t_dif * r_bot_dif * e_dif * e_r_dif * d + e_dif * e_t_dif * d;
  const float a_bot_multi_dif = t_dif * a_bot_dif + t_multi_dif * a_bot_dif;
  const float r_bot_multi_dif = t_dif * r_bot_dif * d + e_dif * e_t_dif * r_bot_dif * d;
  const float a_top_multi_dif = e_dif * e_a_dif + r_bot_multi_dif * e_dif * e_a_dif;
  const float r_multi_dif     = e_dif * e_r_dif + r_bot_multi_dif * (t_dif + e_dif * e_t_dif);

  v4f t0 = {t_dir, t_dif, t_multi_dir, t_multi_dif};
  v4f t1 = {r_bot_multi_dir, r_bot_multi_dif, a_top_multi_dir, a_top_multi_dif};
  o0 = t0;
  o1 = t1;

  r_bot_dir = r_multi_dir;
  r_bot_dif = r_multi_dif;
  a_bot_dir = a_top_multi_dir + a_bot_multi_dir;
  a_bot_dif = a_top_multi_dif + a_bot_multi_dif;
}

__global__ __launch_bounds__(WAVES_PER_BLOCK * 32)
void upward_prop_scan_kernel(const float* __restrict__ x,
                             const float* __restrict__ init_state,
                             float* __restrict__ out,
                             int B, int L) {
  __shared__ float lds[WAVES_PER_BLOCK * RING_D * STEP_FLOATS];

  const int lane = threadIdx.x & 31;
  const int wave = threadIdx.x >> 5;
  const int b    = blockIdx.x * WAVES_PER_BLOCK + wave;
  if (b >= B) return;  // uniform per wave

  float* ring = lds + wave * (RING_D * STEP_FLOATS);
  const float* xb = x   + (size_t)b * L * STEP_FLOATS;
  float*       ob = out + (size_t)b * L * STEP_FLOATS;

  // init state: (B, C, 4) -> lane c reads 4 contiguous floats (dense 16B/lane)
  v4f s = *(const v4f*)(init_state + (size_t)b * 128 + lane * 4);
  float r_bot_dir = s.x, r_bot_dif = s.y, a_bot_dir = s.z, a_bot_dif = s.w;

  // ---- prologue: fill the async ring (dense 16B-per-lane copies) ----------
  const int pre = (L < RING_D) ? L : RING_D;
  for (int p = 0; p < pre; ++p) {
    const float* g  = xb + (size_t)p * STEP_FLOATS;
    float*       lp = ring + p * STEP_FLOATS;
    async_copy_b128(g + lane * 4,       lp + lane * 4);
    async_copy_b128(g + 128 + lane * 4, lp + 128 + lane * 4);
  }

  int l = 0, slot = 0;

  // ---- main loop: steady-state software pipeline --------------------------
  for (; l < L - RING_D; ++l) {
    WAIT_ASYNCCNT(2 * (RING_D - 1));          // oldest step (l) resident in LDS

    const float* mp = ring + slot * STEP_FLOATS + lane * 8;
    v4f i0 = *(const v4f*)(mp);
    v4f i1 = *(const v4f*)(mp + 4);
    WAIT_DSCNT0();                            // slot data safely in VGPRs

    // refill this slot with step l + RING_D
    {
      const float* g  = xb + (size_t)(l + RING_D) * STEP_FLOATS;
      float*       lp = ring + slot * STEP_FLOATS;
      async_copy_b128(g + lane * 4,       lp + lane * 4);
      async_copy_b128(g + 128 + lane * 4, lp + 128 + lane * 4);
    }

    v4f o0, o1;
    cell_step(i0, i1, r_bot_dir, r_bot_dif, a_bot_dir, a_bot_dif, o0, o1);

    float* op = ob + (size_t)l * STEP_FLOATS + lane * 8;
    __builtin_nontemporal_store(o0, (v4f*)op);        // streamed once, skip L2 residency
    __builtin_nontemporal_store(o1, (v4f*)(op + 4));

    slot = (slot + 1 == RING_D) ? 0 : slot + 1;
  }

  // ---- tail: everything already in flight; drain and finish ---------------
  WAIT_ASYNCCNT(0);
  for (; l < L; ++l) {
    const float* mp = ring + slot * STEP_FLOATS + lane * 8;
    v4f i0 = *(const v4f*)(mp);
    v4f i1 = *(const v4f*)(mp + 4);

    v4f o0, o1;
    cell_step(i0, i1, r_bot_dir, r_bot_dif, a_bot_dir, a_bot_dif, o0, o1);

    float* op = ob + (size_t)l * STEP_FLOATS + lane * 8;
    __builtin_nontemporal_store(o0, (v4f*)op);
    __builtin_nontemporal_store(o1, (v4f*)(op + 4));

    slot = (slot + 1 == RING_D) ? 0 : slot + 1;
  }
}

extern "C" void kernel_launch(void* const* d_in, const int* in_sizes, int n_in,
                              void* d_out, int out_size, void* d_ws, size_t ws_size,
                              hipStream_t stream) {
  const float* x          = (const float*)d_in[0];
  const float* init_state = (const float*)d_in[1];
  float*       out        = (float*)d_out;

  const int B = in_sizes[1] / 128;               // init_state is (B, C*4), C = 32
  const int L = in_sizes[0] / (B * STEP_FLOATS); // x is (B, L, C, 8)

  const int blocks = (B + WAVES_PER_BLOCK - 1) / WAVES_PER_BLOCK;
  hipLaunchKernelGGL(upward_prop_scan_kernel, dim3(blocks),
                     dim3(WAVES_PER_BLOCK * 32), 0, stream,
                     x, init_state, out, B, L);
}